// GATv2_34754875359295
// MI455X (gfx1250) — compile-verified
//
#include <hip/hip_runtime.h>
#include <cstdint>
#include <cstddef>

// ---------------- problem constants (from reference) ----------------
#define NNODES   50000
#define INDIM    128
#define NHEADS   2
#define CDIM     64
#define HCDIM    128          // NHEADS*CDIM
#define NEDGES   800000
#define ETOTAL   (NEDGES + NNODES)   // with self loops
#define NEGSLOPE 0.2f
#define EPSGN    1e-5f

typedef float    v2f __attribute__((ext_vector_type(2)));
typedef float    v8f __attribute__((ext_vector_type(8)));
typedef unsigned su4 __attribute__((ext_vector_type(4)));
typedef unsigned su8 __attribute__((ext_vector_type(8)));

__device__ __forceinline__ float lrelu(float v) { return v > 0.f ? v : NEGSLOPE * v; }

__device__ __forceinline__ void atomicMaxF32(float* addr, float val) {
    unsigned int* ua = (unsigned int*)addr;
    unsigned int old = *ua;
    while (__uint_as_float(old) < val) {
        unsigned int assumed = old;
        old = atomicCAS(ua, assumed, __float_as_uint(val));
        if (old == assumed) break;
    }
}

// ---------------- weight transpose: Wt[c*K + k] = W[k*C + c] ----------------
// Makes WMMA B fragments (two consecutive K values of one column) a single b64 load.
__global__ void transposeW(const float* __restrict__ W, float* __restrict__ Wt,
                           int K, int cols) {
    int idx = blockIdx.x * blockDim.x + threadIdx.x;
    if (idx >= K * cols) return;
    int k = idx / cols, c = idx % cols;
    Wt[(size_t)c * K + k] = W[idx];
}

// ---------------- WMMA dual GEMM: Yl = X@Wl+bl, Yr = X@Wr+br ----------------
// One block (8 wave32) per 16-row slab of X; waves cover the 8 column tiles and
// all share one 16x128 A slab staged into LDS by the Tensor Data Mover (TDM).
// f32 WMMA: V_WMMA_F32_16X16X4_F32, A frag = 2 VGPRs (lanes 0-15 K={0,1},
// lanes 16-31 K={2,3}); one A fragment feeds BOTH (Wl, Wr) accumulators.
__global__ void linear2_wmma(const float* __restrict__ X,
                             const float* __restrict__ Wlt, const float* __restrict__ bl,
                             const float* __restrict__ Wrt, const float* __restrict__ br,
                             float* __restrict__ Yl, float* __restrict__ Yr) {
    __shared__ float lds_A[16 * HCDIM];          // 8 KB slab, shared by all 8 waves
    const int wave = threadIdx.x >> 5;
    const int lane = threadIdx.x & 31;
    const int mt = blockIdx.x;

    if (wave == 0) {
        // Build Tensor DMA Descriptor (D#): 1-D tile of 16*128 f32 (contiguous 8 KB).
        unsigned long long ga =
            (unsigned long long)(const void*)(X + (size_t)mt * 16 * HCDIM);
        unsigned ldsOff = (unsigned)(unsigned long long)(const void*)&lds_A[0];
        const unsigned nelem = 16u * HCDIM;      // 2048 elements
        su4 g0;
        g0[0] = 1u;                                          // count=1 (valid D#)
        g0[1] = ldsOff;                                      // lds_addr (bytes)
        g0[2] = (unsigned)(ga & 0xffffffffull);              // global_addr[31:0]
        g0[3] = (unsigned)((ga >> 32) & 0x01ffffffull)       // global_addr[56:32]
              | 0x80000000u;                                 // type=2 ("image")
        su8 g1;
        g1[0] = 0x00020000u;                                 // data_size=4B
        g1[1] = (nelem & 0xffffu) << 16;                     // tensor_dim0[15:0]
        g1[2] = (nelem >> 16) | (1u << 16);                  // dim0[31:16] | tensor_dim1=1
        g1[3] = (nelem & 0xffffu) << 16;                     // tile_dim0 = 2048
        g1[4] = 1u;                                          // tile_dim1=1, tile_dim2=0
        g1[5] = nelem;                                       // tensor_dim0_stride[31:0]
        g1[6] = 0u;
        g1[7] = 0u;
        asm volatile("tensor_load_to_lds %0, %1" :: "s"(g0), "s"(g1) : "memory");
        __builtin_amdgcn_s_wait_tensorcnt(0);                // TENSORcnt -> 0
    }
    __syncthreads();

    const int nt   = wave;                        // column tile (HCDIM/16 == 8 waves)
    const int col  = nt * 16 + (lane & 15);
    const int koff = (lane >> 4) << 1;            // 0 for lanes 0-15, 2 for 16-31
    const float* la  = lds_A + (size_t)(lane & 15) * HCDIM;   // A row in LDS
    const float* wl  = Wlt + (size_t)col * HCDIM;             // transposed: column-major
    const float* wr  = Wrt + (size_t)col * HCDIM;

    v8f cl = {};
    v8f cr = {};
    for (int k = 0; k < HCDIM; k += 4) {
        v2f a;                                    // ds_load_b64
        a.x = la[k + koff];
        a.y = la[k + koff + 1];
        v2f b0;                                   // global_load_b64
        b0.x = wl[k + koff];
        b0.y = wl[k + koff + 1];
        cl = __builtin_amdgcn_wmma_f32_16x16x4_f32(false, a, false, b0,
                                                   (short)0, cl, false, false);
        v2f b1;
        b1.x = wr[k + koff];
        b1.y = wr[k + koff + 1];
        cr = __builtin_amdgcn_wmma_f32_16x16x4_f32(false, a, false, b1,
                                                   (short)0, cr, false, false);
    }
    float biasl = bl[col], biasr = br[col];
#pragma unroll
    for (int r = 0; r < 8; ++r) {
        int m = mt * 16 + ((lane < 16) ? r : r + 8);   // C layout: vgpr r -> row r / r+8
        Yl[(size_t)m * HCDIM + col] = cl[r] + biasl;
        Yr[(size_t)m * HCDIM + col] = cr[r] + biasr;
    }
}

// ---------------- WMMA single GEMM + ReLU (MLP 128->64, transposed W) --------
__global__ void linear_relu_wmma(const float* __restrict__ X,
                                 const float* __restrict__ Wt, const float* __restrict__ b,
                                 float* __restrict__ Y,
                                 int K, int ncols, int nTiles) {
    int gid  = blockIdx.x * blockDim.x + threadIdx.x;
    int wv   = gid >> 5;
    int lane = threadIdx.x & 31;
    if (wv >= nTiles) return;                 // wave-uniform: EXEC stays all-ones
    int colTiles = ncols >> 4;
    int mt = wv / colTiles, nt = wv % colTiles;
    int row  = mt * 16 + (lane & 15);
    int col  = nt * 16 + (lane & 15);
    int koff = (lane >> 4) << 1;
    const float* xrow = X + (size_t)row * K;
    const float* wcol = Wt + (size_t)col * K;

    v8f c = {};
    for (int k = 0; k < K; k += 4) {
        v2f a;
        a.x = xrow[k + koff];
        a.y = xrow[k + koff + 1];
        v2f bb;
        bb.x = wcol[k + koff];
        bb.y = wcol[k + koff + 1];
        c = __builtin_amdgcn_wmma_f32_16x16x4_f32(false, a, false, bb,
                                                  (short)0, c, false, false);
    }
    float bias = b[col];
#pragma unroll
    for (int r = 0; r < 8; ++r) {
        int m = mt * 16 + ((lane < 16) ? r : r + 8);
        Y[(size_t)m * ncols + col] = fmaxf(c[r] + bias, 0.f);
    }
}

// ---------------- init helpers ----------------
__global__ void fill_f32(float* p, float v, size_t n) {
    size_t i = (size_t)blockIdx.x * blockDim.x + threadIdx.x;
    size_t st = (size_t)gridDim.x * blockDim.x;
    for (; i < n; i += st) p[i] = v;
}

__global__ void bias_bcast(float* p, const float* __restrict__ bias, size_t n) {
    size_t i = (size_t)blockIdx.x * blockDim.x + threadIdx.x;
    size_t st = (size_t)gridDim.x * blockDim.x;
    for (; i < n; i += st) p[i] = bias[i & (HCDIM - 1)];
}

// ---------------- edge phase (node tables are L2-resident: 25.6MB << 192MB) --
// pass 1: one thread per (edge, head): logit + segment max into emax[dst]
__global__ void edge_score_max(const long long* __restrict__ ei,
                               const float* __restrict__ xl, const float* __restrict__ xr,
                               const float* __restrict__ att,
                               float* __restrict__ esc, float* __restrict__ emax) {
    int idx = blockIdx.x * blockDim.x + threadIdx.x;
    if (idx >= ETOTAL * NHEADS) return;
    int e  = idx >> 1;
    int hh = idx & 1;
    int s, d;
    if (e < NEDGES) { s = (int)ei[e]; d = (int)ei[NEDGES + e]; }
    else            { s = e - NEDGES; d = s; }
    const float4* pl = (const float4*)(xl + (size_t)s * HCDIM + hh * CDIM);
    const float4* pr = (const float4*)(xr + (size_t)d * HCDIM + hh * CDIM);
    const float4* pa = (const float4*)(att + hh * CDIM);
    float acc = 0.f;
#pragma unroll 4
    for (int c4 = 0; c4 < CDIM / 4; ++c4) {
        float4 l4 = pl[c4];
        float4 r4 = pr[c4];
        float4 w4 = pa[c4];
        acc += lrelu(l4.x + r4.x) * w4.x + lrelu(l4.y + r4.y) * w4.y
             + lrelu(l4.z + r4.z) * w4.z + lrelu(l4.w + r4.w) * w4.w;
    }
    esc[(size_t)e * NHEADS + hh] = acc;
    atomicMaxF32(&emax[d * NHEADS + hh], acc);
}

// pass 2: one thread per (edge, head): exp(e-max), segment sum into denom[dst]
__global__ void edge_exp_sum(const long long* __restrict__ ei,
                             float* __restrict__ esc, const float* __restrict__ emax,
                             float* __restrict__ denom) {
    int idx = blockIdx.x * blockDim.x + threadIdx.x;
    if (idx >= ETOTAL * NHEADS) return;
    int e  = idx >> 1;
    int hh = idx & 1;
    int d = (e < NEDGES) ? (int)ei[NEDGES + e] : e - NEDGES;
    float ex = __expf(esc[(size_t)e * NHEADS + hh] - emax[d * NHEADS + hh]);
    esc[(size_t)e * NHEADS + hh] = ex;
    atomicAdd(&denom[d * NHEADS + hh], ex);
}

// pass 3: one thread per (edge, 32-ch chunk): out[dst] += alpha * xl[src]
__global__ void edge_aggregate(const long long* __restrict__ ei,
                               const float* __restrict__ xl,
                               const float* __restrict__ esc, const float* __restrict__ denom,
                               float* __restrict__ acc) {
    int idx = blockIdx.x * blockDim.x + threadIdx.x;
    if (idx >= ETOTAL * 4) return;
    int e = idx >> 2;
    int q = idx & 3;                          // channel chunk: q*32 .. q*32+31
    int s, d;
    if (e < NEDGES) { s = (int)ei[e]; d = (int)ei[NEDGES + e]; }
    else            { s = e - NEDGES; d = s; }
    int hh = q >> 1;                          // chunk -> head (CDIM==64)
    float al = esc[(size_t)e * NHEADS + hh] / (denom[d * NHEADS + hh] + 1e-16f);
    const float* sp = xl + (size_t)s * HCDIM + q * 32;
    float* dp = acc + (size_t)d * HCDIM + q * 32;
#pragma unroll 8
    for (int j = 0; j < 32; ++j)
        atomicAdd(&dp[j], sp[j] * al);
}

// ---------------- GraphNorm ----------------
__global__ void col_mean(const float* __restrict__ x, float* __restrict__ mean, int n) {
    int ch = blockIdx.x;
    float s = 0.f;
    for (int i = threadIdx.x; i < n; i += blockDim.x) s += x[(size_t)i * HCDIM + ch];
    __shared__ float red[256];
    red[threadIdx.x] = s;
    __syncthreads();
    for (int off = 128; off > 0; off >>= 1) {
        if ((int)threadIdx.x < off) red[threadIdx.x] += red[threadIdx.x + off];
        __syncthreads();
    }
    if (threadIdx.x == 0) mean[ch] = red[0] / (float)n;
}

__global__ void col_var(const float* __restrict__ x, const float* __restrict__ mean,
                        const float* __restrict__ msc, float* __restrict__ var, int n) {
    int ch = blockIdx.x;
    float mu = msc[ch] * mean[ch];
    float s = 0.f;
    for (int i = threadIdx.x; i < n; i += blockDim.x) {
        float v = x[(size_t)i * HCDIM + ch] - mu;
        s += v * v;
    }
    __shared__ float red[256];
    red[threadIdx.x] = s;
    __syncthreads();
    for (int off = 128; off > 0; off >>= 1) {
        if ((int)threadIdx.x < off) red[threadIdx.x] += red[threadIdx.x + off];
        __syncthreads();
    }
    if (threadIdx.x == 0) var[ch] = red[0] / (float)n;
}

__global__ void norm_relu(const float* __restrict__ x,
                          const float* __restrict__ mean, const float* __restrict__ var,
                          const float* __restrict__ g, const float* __restrict__ be,
                          const float* __restrict__ msc,
                          float* __restrict__ out, size_t total) {
    size_t i = (size_t)blockIdx.x * blockDim.x + threadIdx.x;
    size_t st = (size_t)gridDim.x * blockDim.x;
    for (; i < total; i += st) {
        int ch = (int)(i & (HCDIM - 1));
        float v = x[i] - msc[ch] * mean[ch];
        float r = g[ch] * v * rsqrtf(var[ch] + EPSGN) + be[ch];
        out[i] = fmaxf(r, 0.f);
    }
}

// ---------------- MLP tail 64 -> 2 ----------------
__global__ void mlp2(const float* __restrict__ hid, const float* __restrict__ W2,
                     const float* __restrict__ b2, float* __restrict__ out, int n) {
    int i = blockIdx.x * blockDim.x + threadIdx.x;
    if (i >= n) return;
    const float* hp = hid + (size_t)i * CDIM;
    float a0 = b2[0], a1 = b2[1];
#pragma unroll 8
    for (int k = 0; k < CDIM; ++k) {
        float v = hp[k];
        a0 += v * W2[k * 2 + 0];
        a1 += v * W2[k * 2 + 1];
    }
    out[i * 2 + 0] = a0;
    out[i * 2 + 1] = a1;
}

// ---------------- host driver ----------------
extern "C" void kernel_launch(void* const* d_in, const int* in_sizes, int n_in,
                              void* d_out, int out_size, void* d_ws, size_t ws_size,
                              hipStream_t stream) {
    (void)in_sizes; (void)n_in; (void)out_size; (void)ws_size;
    const float*     x    = (const float*)d_in[0];
    const long long* ei   = (const long long*)d_in[1];   // reference dtype: int64
    const float *Wl0 = (const float*)d_in[2],  *bl0 = (const float*)d_in[3];
    const float *Wr0 = (const float*)d_in[4],  *br0 = (const float*)d_in[5];
    const float *att0= (const float*)d_in[6],  *bias0=(const float*)d_in[7];
    const float *Wl1 = (const float*)d_in[8],  *bl1 = (const float*)d_in[9];
    const float *Wr1 = (const float*)d_in[10], *br1 = (const float*)d_in[11];
    const float *att1= (const float*)d_in[12], *bias1=(const float*)d_in[13];
    const float *g0  = (const float*)d_in[14], *be0 = (const float*)d_in[15], *ms0 = (const float*)d_in[16];
    const float *g1  = (const float*)d_in[17], *be1 = (const float*)d_in[18], *ms1 = (const float*)d_in[19];
    const float *W1  = (const float*)d_in[20], *b1  = (const float*)d_in[21];
    const float *W2  = (const float*)d_in[22], *b2  = (const float*)d_in[23];
    float* out = (float*)d_out;

    // workspace layout (floats); everything read is written first each call
    float* ws = (float*)d_ws;
    size_t o = 0;
    float* xl    = ws + o; o += (size_t)NNODES * HCDIM;
    float* xr    = ws + o; o += (size_t)NNODES * HCDIM;
    float* acc   = ws + o; o += (size_t)NNODES * HCDIM;
    float* h     = ws + o; o += (size_t)NNODES * HCDIM;
    float* esc   = ws + o; o += (size_t)ETOTAL * NHEADS;
    float* emax  = ws + o; o += (size_t)NNODES * NHEADS;
    float* denom = ws + o; o += (size_t)NNODES * NHEADS;
    float* mean  = ws + o; o += HCDIM;
    float* var   = ws + o; o += HCDIM;
    float* hid   = ws + o; o += (size_t)NNODES * CDIM;
    float* Wlt   = ws + o; o += (size_t)HCDIM * HCDIM;
    float* Wrt   = ws + o; o += (size_t)HCDIM * HCDIM;
    float* W1t   = ws + o; o += (size_t)HCDIM * CDIM;

    const int ehblocks = (ETOTAL * NHEADS + 255) / 256;
    const int eqblocks = (ETOTAL * 4 + 255) / 256;
    const int nhblocks = (NNODES * NHEADS + 255) / 256;
    const int fullblk  = (int)(((size_t)NNODES * HCDIM + 255) / 256);
    const int wtblk    = (HCDIM * HCDIM + 255) / 256;

    auto conv = [&](const float* X, const float* Wl, const float* bl,
                    const float* Wr, const float* br, const float* att,
                    const float* bias, const float* gg, const float* bb,
                    const float* msc, float* outbuf) {
        transposeW<<<wtblk, 256, 0, stream>>>(Wl, Wlt, HCDIM, HCDIM);
        transposeW<<<wtblk, 256, 0, stream>>>(Wr, Wrt, HCDIM, HCDIM);
        linear2_wmma<<<NNODES / 16, 256, 0, stream>>>(X, Wlt, bl, Wrt, br, xl, xr);
        fill_f32<<<nhblocks, 256, 0, stream>>>(emax, -3.4e38f, (size_t)NNODES * NHEADS);
        fill_f32<<<nhblocks, 256, 0, stream>>>(denom, 0.f, (size_t)NNODES * NHEADS);
        bias_bcast<<<fullblk, 256, 0, stream>>>(acc, bias, (size_t)NNODES * HCDIM);
        edge_score_max<<<ehblocks, 256, 0, stream>>>(ei, xl, xr, att, esc, emax);
        edge_exp_sum<<<ehblocks, 256, 0, stream>>>(ei, esc, emax, denom);
        edge_aggregate<<<eqblocks, 256, 0, stream>>>(ei, xl, esc, denom, acc);
        col_mean<<<HCDIM, 256, 0, stream>>>(acc, mean, NNODES);
        col_var<<<HCDIM, 256, 0, stream>>>(acc, mean, msc, var, NNODES);
        norm_relu<<<fullblk, 256, 0, stream>>>(acc, mean, var, gg, bb, msc, outbuf,
                                               (size_t)NNODES * HCDIM);
    };

    // layer 0: x -> h    (IN == HC == 128, same GEMM shape both layers)
    conv(x, Wl0, bl0, Wr0, br0, att0, bias0, g0, be0, ms0, h);
    // layer 1: h -> acc  (normalize in place)
    conv(h, Wl1, bl1, Wr1, br1, att1, bias1, g1, be1, ms1, acc);

    // MLP head
    transposeW<<<(HCDIM * CDIM + 255) / 256, 256, 0, stream>>>(W1, W1t, HCDIM, CDIM);
    int tiles1 = (NNODES / 16) * (CDIM / 16);
    linear_relu_wmma<<<(tiles1 * 32 + 255) / 256, 256, 0, stream>>>(
        acc, W1t, b1, hid, HCDIM, CDIM, tiles1);
    mlp2<<<(NNODES + 255) / 256, 256, 0, stream>>>(hid, W2, b2, out, NNODES);
}